// NeuralNetwork_11029476016653
// MI455X (gfx1250) — compile-verified
//
#include <hip/hip_runtime.h>

typedef __attribute__((ext_vector_type(2))) float v2f;
typedef __attribute__((ext_vector_type(8))) float v8f;

__device__ __forceinline__ float elu_f(float x) {
    return x > 0.0f ? x : (__expf(x) - 1.0f);
}

// Fused 2-layer MLP:
//   h[b][m]  = elu( dot(x[b][4m:4m+4], w1) + b1 )      m = 0..4
//   out[b]   = elu( dot(h[b][:], w2) )
// First layer done with V_WMMA_F32_16X16X4_F32 (K=4 == window width).
__global__ void __launch_bounds__(256)
fused_mlp_wmma(const float* __restrict__ x,
               const float* __restrict__ w1,
               const float* __restrict__ b1,
               const float* __restrict__ w2,
               float* __restrict__ out,
               int n_rows)
{
    const int lane = threadIdx.x & 31;
    const int wave = threadIdx.x >> 5;
    const long tile = (long)blockIdx.x * 8 + wave;   // 16 rows per wave-tile
    const long base = tile * 16;
    if (base >= n_rows) return;                      // wave-uniform: EXEC stays all-1s

    const int sub = lane & 15;                       // row within tile this lane feeds to A
    const int hi  = (lane >> 4) & 1;                 // 0: K=0..1 (D rows 0..7), 1: K=2..3 (D rows 8..15)

    // ---- A-matrix source row (clamped arithmetically to keep EXEC full) ----
    long arow = base + sub;
    if (arow >= n_rows) arow = n_rows - 1;
    const float* xr = x + arow * 20 + hi * 2;        // 8-byte aligned (80B rows, 16B windows)

    // ---- B-matrix: B[k][n] = w1[k] broadcast across all 16 columns ----
    v2f bm;
    bm.x = hi ? w1[2] : w1[0];                       // VGPR0: K=0 row / K=2 row
    bm.y = hi ? w1[3] : w1[1];                       // VGPR1: K=1 row / K=3 row

    const v8f c0 = {0.f, 0.f, 0.f, 0.f, 0.f, 0.f, 0.f, 0.f};

    // ---- 5 windows -> 5 WMMAs; each D lane then holds dot(row, w1) for 8 rows ----
    v8f d[5];
#pragma unroll
    for (int m = 0; m < 5; ++m) {
        v2f a = *(const v2f*)(xr + m * 4);           // global_load_b64
        d[m] = __builtin_amdgcn_wmma_f32_16x16x4_f32(
                   /*neg_a=*/false, a, /*neg_b=*/false, bm,
                   /*c_mod=*/(short)0, c0, /*reuse_a=*/false, /*reuse_b=*/false);
    }

    // ---- Epilogue (SIMD-redundant across N lanes; costs nothing extra) ----
    const float bias = b1[0];
    float w2v[5];
#pragma unroll
    for (int m = 0; m < 5; ++m) w2v[m] = w2[m];

    float o[8];
#pragma unroll
    for (int r = 0; r < 8; ++r) {
        float acc = 0.0f;
#pragma unroll
        for (int m = 0; m < 5; ++m)
            acc += elu_f(d[m][r] + bias) * w2v[m];
        o[r] = elu_f(acc);
    }

    // ---- Lane 0 writes D rows 0..7, lane 16 writes rows 8..15 ----
    if (sub == 0) {
        const long obase = base + hi * 8;
#pragma unroll
        for (int r = 0; r < 8; ++r) {
            long row = obase + r;
            if (row < n_rows) out[row] = o[r];
        }
    }
}

extern "C" void kernel_launch(void* const* d_in, const int* in_sizes, int n_in,
                              void* d_out, int out_size, void* d_ws, size_t ws_size,
                              hipStream_t stream) {
    const float* x  = (const float*)d_in[0];   // (B, 20) f32
    const float* w1 = (const float*)d_in[1];   // (4,)
    const float* b1 = (const float*)d_in[2];   // (1,)
    const float* w2 = (const float*)d_in[3];   // (5,)
    float* out = (float*)d_out;                // (B, 1) f32

    const int n_rows = in_sizes[0] / 20;
    const int tiles  = (n_rows + 15) / 16;     // 16 rows per wave
    const int blocks = (tiles + 7) / 8;        // 8 waves (256 threads) per block

    fused_mlp_wmma<<<blocks, 256, 0, stream>>>(x, w1, b1, w2, out, n_rows);
}